// GatbertEmbeddings_28879360099207
// MI455X (gfx1250) — compile-verified
//
#include <hip/hip_runtime.h>
#include <hip/hip_bf16.h>

// Shapes (compile-time constants from the reference)
#define BATCH   32
#define SEQ     512
#define HID     768
#define NODES   256
#define NNZ_    65536
#define LN_EPS  1e-12f

typedef __attribute__((ext_vector_type(2))) float v2f;
typedef __attribute__((ext_vector_type(8))) float v8f;

// ---------------------------------------------------------------------------
// Kernel 1: embedding gather-sum.  sub[row, :] = word[ids] + pos[pids] + type[tids]
// One block per (b,s) row; 192 threads x float4 = 768 floats. Fully coalesced.
// ---------------------------------------------------------------------------
__global__ void embed_sum_kernel(const int* __restrict__ ids,
                                 const int* __restrict__ pids,
                                 const int* __restrict__ tids,
                                 const float* __restrict__ wemb,
                                 const float* __restrict__ pemb,
                                 const float* __restrict__ temb,
                                 float* __restrict__ sub) {
    const int row = blockIdx.x;          // 0 .. B*S-1
    const int t   = threadIdx.x;         // 0 .. 191
    const int id = ids[row];
    const int pi = pids[row];
    const int ty = tids[row];
    const float4 a = ((const float4*)(wemb + (size_t)id * HID))[t];
    const float4 b = ((const float4*)(pemb + (size_t)pi * HID))[t];
    const float4 c = ((const float4*)(temb + (size_t)ty * HID))[t];
    float4 r;
    r.x = a.x + b.x + c.x;
    r.y = a.y + b.y + c.y;
    r.z = a.z + b.z + c.z;
    r.w = a.w + b.w + c.w;
    ((float4*)(sub + (size_t)row * HID))[t] = r;
}

// ---------------------------------------------------------------------------
// Kernel 2: densify the sparse pooling mask.  mask[b][n][s] += v
// atomicAdd handles duplicate (b,n,s) triples exactly like segment_sum.
// ---------------------------------------------------------------------------
__global__ void scatter_mask_kernel(const int* __restrict__ mb,
                                    const int* __restrict__ mn,
                                    const int* __restrict__ ms,
                                    const float* __restrict__ mv,
                                    float* __restrict__ mask) {
    const int e = blockIdx.x * blockDim.x + threadIdx.x;
    if (e < NNZ_) {
        const int b = mb[e], n = mn[e], s = ms[e];
        atomicAdd(mask + ((size_t)b * NODES + n) * SEQ + s, mv[e]);
    }
}

// ---------------------------------------------------------------------------
// Kernel 3: per-batch dense GEMM via V_WMMA_F32_16X16X4_F32 (exact fp32).
//   out[b][256x768] = mask[b][256x512] @ sub[b][512x768]
//
// 2x2 register blocking: each wave owns a 32x32 C block (four 16x16 tiles,
// 4 x v8f accumulators). Per K-step: 2 A-fragment loads + 2 B-fragment loads
// feed 4 WMMAs -> 8 B/lane/wmma (vs 16 unblocked), halving L1/L2 fragment
// traffic on this bandwidth-bound problem. EXEC is all ones (WMMA req).
//
// VGPR layouts (ISA 7.12.2, 32-bit):
//   A 16x4 : lanes 0-15 -> M=lane, K={k0,k0+1}; lanes 16-31 -> K={k0+2,k0+3}
//   B 4x16 : lane holds col n0+(lane&15); VGPR0/1 hold rows K=k0+koff{,+1}
//   C 16x16: row = m0 + vgpr + 8*(lane>=16), col = n0 + (lane&15)
// ---------------------------------------------------------------------------
__global__ __launch_bounds__(256) void pool_gemm_wmma_kernel(
        const float* __restrict__ mask,   // [B, NODES, SEQ]
        const float* __restrict__ sub,    // [B, SEQ, HID]
        float* __restrict__ out) {        // [B, NODES, HID]  (pre-LN)
    const int lane   = threadIdx.x;                 // 0..31
    const int wave   = threadIdx.y;                 // 0..7
    const int ngroup = blockIdx.x * 8 + wave;       // 0..23 (each = 2 n-tiles)
    const int mgroup = blockIdx.y;                  // 0..7  (each = 2 m-tiles)
    const int b      = blockIdx.z;                  // 0..31

    const int n0   = ngroup * 32;
    const int m0   = mgroup * 32;
    const int l15  = lane & 15;
    const int half = lane >> 4;                     // 0 or 1
    const int koff = half * 2;

    const float* A0 = mask + ((size_t)b * NODES + m0 + l15) * SEQ;       // m-tile 0 row
    const float* A1 = A0 + (size_t)16 * SEQ;                             // m-tile 1 row
    const float* Bb = sub  + (size_t)b * SEQ * HID + n0 + l15;           // n-tile 0 col

    v8f c00 = {0.f,0.f,0.f,0.f,0.f,0.f,0.f,0.f};
    v8f c01 = c00, c10 = c00, c11 = c00;

    for (int k0 = 0; k0 < SEQ; k0 += 4) {
        v2f a0, a1, b0, b1;
        // A fragments: two consecutive K values for this lane's half
        a0.x = A0[k0 + koff];
        a0.y = A0[k0 + koff + 1];
        a1.x = A1[k0 + koff];
        a1.y = A1[k0 + koff + 1];
        // B fragments: rows K = k0+koff, k0+koff+1; cols n0+l15 and n0+16+l15
        const float* bp = Bb + (size_t)(k0 + koff) * HID;
        b0.x = bp[0];
        b0.y = bp[HID];
        b1.x = bp[16];
        b1.y = bp[HID + 16];
        // 8 args: (neg_a, A, neg_b, B, c_mod, C, reuse_a, reuse_b)
        c00 = __builtin_amdgcn_wmma_f32_16x16x4_f32(false, a0, false, b0, (short)0, c00, false, false);
        c01 = __builtin_amdgcn_wmma_f32_16x16x4_f32(false, a0, false, b1, (short)0, c01, false, false);
        c10 = __builtin_amdgcn_wmma_f32_16x16x4_f32(false, a1, false, b0, (short)0, c10, false, false);
        c11 = __builtin_amdgcn_wmma_f32_16x16x4_f32(false, a1, false, b1, (short)0, c11, false, false);
    }

    // store: tile (i,j): row = m0 + 16*i + vgpr + 8*half, col = n0 + 16*j + l15
    float* o00 = out + ((size_t)b * NODES + m0 + half * 8) * HID + n0 + l15;
    float* o10 = o00 + (size_t)16 * HID;
    #pragma unroll
    for (int r = 0; r < 8; ++r) {
        o00[(size_t)r * HID]      = c00[r];
        o00[(size_t)r * HID + 16] = c01[r];
        o10[(size_t)r * HID]      = c10[r];
        o10[(size_t)r * HID + 16] = c11[r];
    }
}

// ---------------------------------------------------------------------------
// Kernel 4: in-place LayerNorm over H=768. One 256-thread block (8 wave32)
// per node row; wave shuffle reduce + LDS cross-wave combine.
// ---------------------------------------------------------------------------
__global__ __launch_bounds__(256) void layernorm_kernel(
        float* __restrict__ out,
        const float* __restrict__ gamma,
        const float* __restrict__ beta) {
    const int row = blockIdx.x;                 // 0 .. B*NODES-1
    const int t   = threadIdx.x;                // 0 .. 255
    float* p = out + (size_t)row * HID;

    const float x0 = p[t];
    const float x1 = p[t + 256];
    const float x2 = p[t + 512];
    float s  = x0 + x1 + x2;
    float sq = x0 * x0 + x1 * x1 + x2 * x2;

    // wave32 reduce
    #pragma unroll
    for (int off = 16; off > 0; off >>= 1) {
        s  += __shfl_down(s,  off, 32);
        sq += __shfl_down(sq, off, 32);
    }

    __shared__ float ssum[8], ssq[8];
    __shared__ float s_mu, s_rstd;
    const int wid = t >> 5;
    if ((t & 31) == 0) { ssum[wid] = s; ssq[wid] = sq; }
    __syncthreads();
    if (t == 0) {
        float S = 0.f, Q = 0.f;
        #pragma unroll
        for (int i = 0; i < 8; ++i) { S += ssum[i]; Q += ssq[i]; }
        const float mu  = S * (1.0f / HID);
        const float var = Q * (1.0f / HID) - mu * mu;
        s_mu   = mu;
        s_rstd = rsqrtf(var + LN_EPS);
    }
    __syncthreads();
    const float mu = s_mu, rs = s_rstd;

    p[t]       = (x0 - mu) * rs * gamma[t]       + beta[t];
    p[t + 256] = (x1 - mu) * rs * gamma[t + 256] + beta[t + 256];
    p[t + 512] = (x2 - mu) * rs * gamma[t + 512] + beta[t + 512];
}

// ---------------------------------------------------------------------------
// Launch
// ---------------------------------------------------------------------------
extern "C" void kernel_launch(void* const* d_in, const int* in_sizes, int n_in,
                              void* d_out, int out_size, void* d_ws, size_t ws_size,
                              hipStream_t stream) {
    const int*   input_ids      = (const int*)  d_in[0];
    const int*   position_ids   = (const int*)  d_in[1];
    const int*   token_type_ids = (const int*)  d_in[2];
    const int*   mask_b         = (const int*)  d_in[3];
    const int*   mask_n         = (const int*)  d_in[4];
    const int*   mask_s         = (const int*)  d_in[5];
    const float* mask_values    = (const float*)d_in[6];
    // d_in[7] = max_nodes scalar (compile-time NODES here)
    const float* word_emb       = (const float*)d_in[8];
    const float* pos_emb        = (const float*)d_in[9];
    const float* type_emb       = (const float*)d_in[10];
    const float* ln_gamma       = (const float*)d_in[11];
    const float* ln_beta        = (const float*)d_in[12];
    float* out = (float*)d_out;

    // workspace layout
    const size_t sub_bytes  = (size_t)BATCH * SEQ * HID * sizeof(float);    // 50,331,648
    const size_t mask_bytes = (size_t)BATCH * NODES * SEQ * sizeof(float);  // 16,777,216
    float* sub   = (float*)d_ws;
    float* maskd = (float*)((char*)d_ws + sub_bytes);
    (void)in_sizes; (void)n_in; (void)out_size; (void)ws_size;

    // zero the dense mask each call (graph-capture-safe)
    hipMemsetAsync(maskd, 0, mask_bytes, stream);

    // 1) embedding gather-sum -> sub [B*S, H]
    embed_sum_kernel<<<BATCH * SEQ, HID / 4, 0, stream>>>(
        input_ids, position_ids, token_type_ids,
        word_emb, pos_emb, type_emb, sub);

    // 2) densify sparse mask
    scatter_mask_kernel<<<NNZ_ / 256, 256, 0, stream>>>(
        mask_b, mask_n, mask_s, mask_values, maskd);

    // 3) per-batch GEMM via fp32 WMMA (2x2 register-blocked) -> out (pre-LN)
    dim3 gg(3, 8, BATCH);   // 24 n-groups (8 waves/block), 8 m-groups, 32 batches
    dim3 gb(32, 8);         // 8 waves, each a 32x32 C block
    pool_gemm_wmma_kernel<<<gg, gb, 0, stream>>>(maskd, sub, out);

    // 4) in-place LayerNorm
    layernorm_kernel<<<BATCH * NODES, 256, 0, stream>>>(out, ln_gamma, ln_beta);
}